// WriteGateMemory_35270271435209
// MI455X (gfx1250) — compile-verified
//
#include <hip/hip_runtime.h>
#include <hip/hip_bf16.h>

#define B_   32
#define T_   4096
#define H_   512
#define V_   64
#define KW_  32
#define MSL_ 64
#define RSQRT_H 0.044194173824159216f  // 1/sqrt(512)

typedef __attribute__((ext_vector_type(2))) float v2f;
typedef __attribute__((ext_vector_type(8))) float v8f;

// ---------------------------------------------------------------------------
// Kernel 1: gate scores.  gate[b,t] = sigmoid(enc[b,t,:]·Wg + bg)
// One wave per token; lane l holds Wg[4l+128j] as float4 in registers.
// 256 blocks (b * 8 chunks) x 256 threads (8 waves), 512 tokens per block.
// Streams all 256 MB of enc_hidden exactly once: the roofline of the problem.
// ---------------------------------------------------------------------------
__global__ void gate_kernel(const float* __restrict__ enc,
                            const float* __restrict__ Wg,
                            const float* __restrict__ bgp,
                            float* __restrict__ gate_out,
                            float* __restrict__ partial) {
    const int b     = blockIdx.x >> 3;
    const int chunk = blockIdx.x & 7;
    const int lane  = threadIdx.x & 31;
    const int wave  = threadIdx.x >> 5;
    const float bg  = bgp[0];

    float4 wg[4];
    const float4* wg4 = (const float4*)Wg;
#pragma unroll
    for (int j = 0; j < 4; ++j) wg[j] = wg4[lane + 32 * j];

    float wsum = 0.f;
    const int base_t = chunk * 512;
    for (int tl = wave; tl < 512; tl += 8) {
        const int t = base_t + tl;
        const float4* row = (const float4*)(enc + ((size_t)b * T_ + t) * H_);
        // prefetch this wave's next token row (global_prefetch_b8)
        if (tl + 8 < 512)
            __builtin_prefetch(enc + ((size_t)b * T_ + t + 8) * H_ + lane * 16, 0, 1);
        float acc = 0.f;
#pragma unroll
        for (int j = 0; j < 4; ++j) {
            float4 e = row[lane + 32 * j];
            acc += e.x * wg[j].x + e.y * wg[j].y + e.z * wg[j].z + e.w * wg[j].w;
        }
#pragma unroll
        for (int off = 16; off > 0; off >>= 1) acc += __shfl_down(acc, off, 32);
        if (lane == 0) {
            float g = 1.f / (1.f + __expf(-(acc + bg)));
            gate_out[b * T_ + t] = g;
            wsum += g;
        }
    }
    __shared__ float sws[8];
    if (lane == 0) sws[wave] = wsum;
    __syncthreads();
    if (threadIdx.x == 0) {
        float s = 0.f;
#pragma unroll
        for (int i = 0; i < 8; ++i) s += sws[i];
        partial[blockIdx.x] = s;
    }
}

// ---------------------------------------------------------------------------
// Kernel 2: write_rate = mean(gate_scores). Deterministic (no float atomics).
// ---------------------------------------------------------------------------
__global__ void writerate_kernel(const float* __restrict__ partial,
                                 float* __restrict__ rate_out) {
    __shared__ float s[256];
    s[threadIdx.x] = partial[threadIdx.x];
    __syncthreads();
    for (int off = 128; off > 0; off >>= 1) {
        if (threadIdx.x < off) s[threadIdx.x] += s[threadIdx.x + off];
        __syncthreads();
    }
    if (threadIdx.x == 0) rate_out[0] = s[0] * (1.f / (float)(B_ * T_));
}

// ---------------------------------------------------------------------------
// Kernel 3: per-batch top-32 of 4096 (descending, smaller index on ties).
// Scores staged in LDS; 32 rounds of parallel argmax (selection sort).
// ---------------------------------------------------------------------------
__global__ void topk_kernel(const float* __restrict__ gate,
                            int* __restrict__ top_idx) {
    __shared__ float vals[T_];
    __shared__ float rv[256];
    __shared__ int   ri[256];
    const int b = blockIdx.x, tid = threadIdx.x;
    for (int i = tid; i < T_; i += 256) vals[i] = gate[b * T_ + i];
    __syncthreads();
    for (int it = 0; it < KW_; ++it) {
        float bestV = -INFINITY; int bestI = 0;
        for (int j = 0; j < T_ / 256; ++j) {
            int i = tid + 256 * j;
            float v = vals[i];
            if (v > bestV || (v == bestV && i < bestI)) { bestV = v; bestI = i; }
        }
        rv[tid] = bestV; ri[tid] = bestI;
        __syncthreads();
        for (int off = 128; off > 0; off >>= 1) {
            if (tid < off) {
                float v2 = rv[tid + off]; int i2 = ri[tid + off];
                if (v2 > rv[tid] || (v2 == rv[tid] && i2 < ri[tid])) { rv[tid] = v2; ri[tid] = i2; }
            }
            __syncthreads();
        }
        if (tid == 0) {
            top_idx[b * KW_ + it] = ri[0];
            vals[ri[0]] = -INFINITY;
        }
        __syncthreads();
    }
}

// ---------------------------------------------------------------------------
// Kernel 4: WMMA f32 GEMM, M=32 fixed (the batch dim), compile-time layout.
//   C[32xN] = A[32xK] * B[KxN] (+ bias[N])
//   BTRANS=1: B[k][n] = W[n*K+k]   (W stored [N x K], i.e. X @ W^T)
//   BTRANS=0: B[k][n] = W[k*N+n]   (W stored [K x N])
// One wave per 16x16 tile; V_WMMA_F32_16X16X4_F32, K stepped 4, unrolled x4
// so the loop body is 4 back-to-back v_wmma with loads hoisted in front.
// ---------------------------------------------------------------------------
template <bool BTRANS, bool HAS_BIAS>
__global__ void wmma_gemm32_kernel(const float* __restrict__ A,
                                   const float* __restrict__ W,
                                   const float* __restrict__ bias,
                                   float* __restrict__ C,
                                   int N, int K) {
    const int wave = threadIdx.x >> 5;
    const int lane = threadIdx.x & 31;
    const int tile = blockIdx.x * 8 + wave;
    const int ntn  = N >> 4;
    const int m0   = (tile / ntn) << 4;
    const int n0   = (tile % ntn) << 4;
    const int col  = lane & 15;
    const int half = lane >> 4;

    v8f c;
    const float cinit = HAS_BIAS ? bias[n0 + col] : 0.f;
#pragma unroll
    for (int r = 0; r < 8; ++r) c[r] = cinit;

    // per-lane base pointers; constant stride per K-step of 4
    const float* ap = A + (m0 + col) * K + (half << 1);
    const float* bp = BTRANS ? (W + (size_t)(n0 + col) * K + (half << 1))
                             : (W + (size_t)(half << 1) * N + (n0 + col));
    const size_t bstep = BTRANS ? 4 : (size_t)4 * N;
    const size_t brow  = BTRANS ? 1 : (size_t)N;

    for (int k0 = 0; k0 < K; k0 += 16) {
        v2f a[4], bv[4];
#pragma unroll
        for (int u = 0; u < 4; ++u) {
            a[u][0]  = ap[4 * u];
            a[u][1]  = ap[4 * u + 1];
            bv[u][0] = bp[(size_t)u * bstep];
            bv[u][1] = bp[(size_t)u * bstep + brow];
        }
#pragma unroll
        for (int u = 0; u < 4; ++u)
            c = __builtin_amdgcn_wmma_f32_16x16x4_f32(false, a[u], false, bv[u],
                                                      (short)0, c, false, false);
        ap += 16;
        bp += 4 * bstep;
    }
#pragma unroll
    for (int r = 0; r < 8; ++r) {
        const int row = m0 + r + (half << 3);
        C[row * N + n0 + col] = c[r];
    }
}

// ---------------------------------------------------------------------------
// Kernel 5: per-batch attention over memory slots (algebraically reduced).
//   s0 = q[b]·bk ; score[m] = (qk[b]·enc[b,idx[m],:] + s0)/sqrt(H)
//   zero slots (32..63) all score s0/sqrt(H) -> softmax denom += 32*exp(z-max)
//   rq[b] = retrieved + query[b]
// ---------------------------------------------------------------------------
__global__ void scores_kernel(const float* __restrict__ enc,
                              const float* __restrict__ query,
                              const float* __restrict__ q,
                              const float* __restrict__ qk,
                              const float* __restrict__ bk,
                              const int* __restrict__ top_idx,
                              float* __restrict__ rq) {
    const int b    = blockIdx.x;
    const int tid  = threadIdx.x;
    const int lane = tid & 31;
    const int wave = tid >> 5;
    __shared__ int   sIdx[KW_];
    __shared__ float sSc[KW_];
    __shared__ float sAttn[KW_];
    if (tid < KW_) sIdx[tid] = top_idx[b * KW_ + tid];
    __syncthreads();

    // s0 = q[b]·bk  (every wave computes it; trivial)
    const float4* q4  = (const float4*)(q + b * H_);
    const float4* bk4 = (const float4*)bk;
    float s0 = 0.f;
#pragma unroll
    for (int j = 0; j < 4; ++j) {
        float4 a = q4[lane + 32 * j], c = bk4[lane + 32 * j];
        s0 += a.x * c.x + a.y * c.y + a.z * c.z + a.w * c.w;
    }
#pragma unroll
    for (int m = 16; m > 0; m >>= 1) s0 += __shfl_xor(s0, m, 32);

    // selected-slot scores
    const float4* qk4 = (const float4*)(qk + b * H_);
    for (int m = wave; m < KW_; m += 8) {
        const float4* row = (const float4*)(enc + ((size_t)b * T_ + sIdx[m]) * H_);
        float acc = 0.f;
#pragma unroll
        for (int j = 0; j < 4; ++j) {
            float4 e = row[lane + 32 * j], k = qk4[lane + 32 * j];
            acc += e.x * k.x + e.y * k.y + e.z * k.z + e.w * k.w;
        }
#pragma unroll
        for (int mm = 16; mm > 0; mm >>= 1) acc += __shfl_xor(acc, mm, 32);
        if (lane == 0) sSc[m] = (acc + s0) * RSQRT_H;
    }
    __syncthreads();

    // softmax over 64 slots (32 real + 32 identical zero-slot scores)
    if (wave == 0) {
        const float z = s0 * RSQRT_H;
        float v  = sSc[lane];
        float mv = v;
#pragma unroll
        for (int m = 16; m > 0; m >>= 1) mv = fmaxf(mv, __shfl_xor(mv, m, 32));
        mv = fmaxf(mv, z);
        float e = __expf(v - mv);
        float ssum = e;
#pragma unroll
        for (int m = 16; m > 0; m >>= 1) ssum += __shfl_xor(ssum, m, 32);
        const float denom = ssum + 32.f * __expf(z - mv);
        sAttn[lane] = e / denom;
    }
    __syncthreads();

    // retrieved + query
    for (int h = tid; h < H_; h += 256) {
        float r = 0.f;
#pragma unroll 4
        for (int m = 0; m < KW_; ++m)
            r += sAttn[m] * enc[((size_t)b * T_ + sIdx[m]) * H_ + h];
        rq[b * H_ + h] = r + query[b * H_ + h];
    }
}

// ---------------------------------------------------------------------------
extern "C" void kernel_launch(void* const* d_in, const int* in_sizes, int n_in,
                              void* d_out, int out_size, void* d_ws, size_t ws_size,
                              hipStream_t stream) {
    const float* enc   = (const float*)d_in[0];
    const float* query = (const float*)d_in[1];
    const float* Wg    = (const float*)d_in[2];
    const float* bg    = (const float*)d_in[3];
    const float* Wq    = (const float*)d_in[4];
    const float* bq    = (const float*)d_in[5];
    const float* Wk    = (const float*)d_in[6];
    const float* bk    = (const float*)d_in[7];
    const float* Wo    = (const float*)d_in[8];
    const float* bo    = (const float*)d_in[9];

    float* out_logits = (float*)d_out;                 // (B,V)   = 2048
    float* out_gate   = out_logits + B_ * V_;          // (B,T)   = 131072
    float* out_rate   = out_gate + B_ * T_;            // scalar

    float* wsf     = (float*)d_ws;
    float* partial = wsf;                              // 256
    float* q       = wsf + 256;                        // 32*512
    float* qk      = q  + B_ * H_;                     // 32*512
    float* rq      = qk + B_ * H_;                     // 32*512
    int*   topIdx  = (int*)(rq + B_ * H_);             // 32*32

    // 1) gate scores (streams all 256 MB of enc_hidden once — the roofline)
    gate_kernel<<<B_ * 8, 256, 0, stream>>>(enc, Wg, bg, out_gate, partial);
    // 2) write_rate
    writerate_kernel<<<1, 256, 0, stream>>>(partial, out_rate);
    // 3) top-32 per batch
    topk_kernel<<<B_, 256, 0, stream>>>(out_gate, topIdx);
    // 4) q = query @ Wq^T + bq           (WMMA, N=512, K=512, B^T)
    wmma_gemm32_kernel<true, true><<<8, 256, 0, stream>>>(query, Wq, bq, q, H_, H_);
    // 5) qk = q @ Wk                     (WMMA, N=512, K=512)
    wmma_gemm32_kernel<false, false><<<8, 256, 0, stream>>>(q, Wk, nullptr, qk, H_, H_);
    // 6) scores/softmax/retrieve -> rq = retrieved + query
    scores_kernel<<<B_, 256, 0, stream>>>(enc, query, q, qk, bk, topIdx, rq);
    // 7) logits = rq @ Wo^T + bo         (WMMA, N=64, K=512, B^T)
    wmma_gemm32_kernel<true, true><<<1, 256, 0, stream>>>(rq, Wo, bo, out_logits, V_, H_);
}